// MOSModel_4810363372446
// MI455X (gfx1250) — compile-verified
//
#include <hip/hip_runtime.h>

// CDNA5 (gfx1250) wave32 WMMA fragment types
typedef __attribute__((ext_vector_type(2))) float v2f;  // A/B frag of V_WMMA_F32_16X16X4_F32
typedef __attribute__((ext_vector_type(8))) float v8f;  // 16x16 f32 C/D accumulator

#ifndef __has_builtin
#define __has_builtin(x) 0
#endif
#define HAVE_WMMA_F32X4 __has_builtin(__builtin_amdgcn_wmma_f32_16x16x4_f32)

// ----------------------------- elementwise kernels (float4-wide) ---------------
__global__ void k_zero4(float4* __restrict__ p, int n4) {
  int i = blockIdx.x * blockDim.x + threadIdx.x;
  if (i < n4) p[i] = make_float4(0.f, 0.f, 0.f, 0.f);
}
__global__ void k_relu4(float4* __restrict__ p, int n4) {
  int i = blockIdx.x * blockDim.x + threadIdx.x;
  if (i >= n4) return;
  float4 v = p[i];
  p[i] = make_float4(fmaxf(v.x, 0.f), fmaxf(v.y, 0.f), fmaxf(v.z, 0.f), fmaxf(v.w, 0.f));
}
__global__ void k_add_relu4(float4* __restrict__ d, const float4* __restrict__ a,
                            const float4* __restrict__ b, int n4) {
  int i = blockIdx.x * blockDim.x + threadIdx.x;
  if (i >= n4) return;
  float4 va = a[i], vb = b[i];
  d[i] = make_float4(fmaxf(va.x + vb.x, 0.f), fmaxf(va.y + vb.y, 0.f),
                     fmaxf(va.z + vb.z, 0.f), fmaxf(va.w + vb.w, 0.f));
}
// cat[v] = [xu[v] (32ch) | x0[v] (8ch)]; one float4 chunk per thread (10 chunks/voxel)
__global__ void k_concat4(float4* __restrict__ cat, const float4* __restrict__ xu,
                          const float4* __restrict__ x0, int n4 /* = M0*10 */) {
  int i = blockIdx.x * blockDim.x + threadIdx.x;
  if (i >= n4) return;
  int v = i / 10, j = i % 10;
  cat[i] = (j < 8) ? xu[v * 8 + j] : x0[v * 2 + (j - 8)];
}

// ----------------------------- stem: 1 -> 8 (K=1, scalar path) -----------------
__global__ void k_stem(const float* __restrict__ feats, const float* __restrict__ W0,
                       const int* __restrict__ pin, const int* __restrict__ pout,
                       const int* __restrict__ kidx, float* __restrict__ out, int P) {
  int p = blockIdx.x * blockDim.x + threadIdx.x;
  if (p >= P) return;
  float f = feats[pin[p]];
  const float* w = W0 + (size_t)kidx[p] * 8;
  float* o = out + (size_t)pout[p] * 8;
#pragma unroll
  for (int c = 0; c < 8; ++c) atomicAdd(&o[c], f * w[c]);
}

// ----------------------------- final: out[p] = z[p2v[p]] @ Wfin (8->3) ---------
__global__ void k_final(const float* __restrict__ z, const float* __restrict__ Wfin,
                        const int* __restrict__ p2v, float* __restrict__ out, int N) {
  int p = blockIdx.x * blockDim.x + threadIdx.x;
  if (p >= N) return;
  const float4* zr = (const float4*)(z + (size_t)p2v[p] * 8);
  float4 z0 = zr[0], z1 = zr[1];
  float zc[8] = {z0.x, z0.y, z0.z, z0.w, z1.x, z1.y, z1.z, z1.w};
  float o0 = 0.f, o1 = 0.f, o2 = 0.f;
#pragma unroll
  for (int c = 0; c < 8; ++c) {
    o0 += zc[c] * Wfin[c * 3 + 0];
    o1 += zc[c] * Wfin[c * 3 + 1];
    o2 += zc[c] * Wfin[c * 3 + 2];
  }
  out[p * 3 + 0] = o0; out[p * 3 + 1] = o1; out[p * 3 + 2] = o2;
}

// ----------------------------- weight repack into B-fragment order -------------
// pW[((k*KSTEPS + kk)*NTILES + t)*32 + lane] = { W[k][c0][n], W[k][c0+1][n] }
// with c0 = kk*4 + (lane<16 ? 0 : 2), n = t*16 + (lane&15); columns >= COUT
// pre-zeroed so B loads need no column masking (only the scatter masks).
template <int CIN, int COUT>
__global__ void k_packW(const float* __restrict__ W, v2f* __restrict__ pW, int total) {
  constexpr int KSTEPS = CIN / 4;
  constexpr int NTILES = (COUT + 15) / 16;
  int i = blockIdx.x * blockDim.x + threadIdx.x;
  if (i >= total) return;
  int lane = i & 31;
  int frag = i >> 5;
  int t = frag % NTILES;
  int kk = (frag / NTILES) % KSTEPS;
  int k = frag / (NTILES * KSTEPS);
  int c0 = kk * 4 + ((lane < 16) ? 0 : 2);
  int n = t * 16 + (lane & 15);
  const float* Wk = W + (size_t)k * CIN * COUT;
  v2f v;
  v.x = (n < COUT) ? Wk[c0 * COUT + n] : 0.f;
  v.y = (n < COUT) ? Wk[(c0 + 1) * COUT + n] : 0.f;
  pW[i] = v;
}

// ----------------------------- gather-WMMA-scatter sparse conv -----------------
// One wave32 per 16 consecutive kernel-map pairs. Pairs are sorted by kernel
// offset k; boundary tiles handled by a distinct-k loop with row-masked A
// fragments (EXEC all-1s across every WMMA). Gathers are unconditional from
// clamped addresses + cndmask; B fragments are single b64 loads from packed
// weights; the scatter has a guard-free full-tile fast path.
#if HAVE_WMMA_F32X4
template <int CIN, int COUT>
__global__ void k_sconv(const float* __restrict__ x, const v2f* __restrict__ pW,
                        const int* __restrict__ pin, const int* __restrict__ pout,
                        const int* __restrict__ kidx, float* __restrict__ out, int P) {
  static_assert(CIN % 4 == 0, "CIN must be a multiple of 4");
  constexpr int KSTEPS = CIN / 4;
  constexpr int NTILES = (COUT + 15) / 16;
  const int lane = threadIdx.x & 31;
  const int wave = blockIdx.x * (blockDim.x >> 5) + (threadIdx.x >> 5);
  const int p0 = wave * 16;
  if (p0 >= P) return;                       // wave-uniform branch
  const int m = lane & 15;                   // row of the 16-pair tile
  const int p = p0 + m;
  const bool valid = p < P;
  const int pc = valid ? p : (P - 1);        // clamp: loads always in-bounds
  const int gi  = pin[pc];
  const int myk = kidx[pc];
  const int koff = (lane < 16) ? 0 : 2;      // which K pair this lane feeds

  v8f acc[NTILES] = {};
  unsigned rem = (unsigned)__ballot(valid) & 0xFFFFu;
  while (rem) {
    const int first = __ffs(rem) - 1;
    const int kcur = __shfl(myk, first, 32);
    rem &= ~((unsigned)__ballot(myk == kcur) & 0xFFFFu);
    const bool inpass = valid && (myk == kcur);
    const v2f* __restrict__ pWk = pW + (size_t)kcur * (KSTEPS * NTILES * 32) + lane;
    const float* __restrict__ xr = x + (size_t)gi * CIN + koff;
#pragma unroll
    for (int kk = 0; kk < KSTEPS; ++kk) {
      const v2f av = *(const v2f*)(xr + kk * 4);   // global_load_b64, unconditional
      v2f a;
      a.x = inpass ? av.x : 0.0f;                  // v_cndmask row mask
      a.y = inpass ? av.y : 0.0f;
#pragma unroll
      for (int t = 0; t < NTILES; ++t) {
        const v2f b = pWk[(kk * NTILES + t) * 32]; // b64 load, base + imm offset
        acc[t] = __builtin_amdgcn_wmma_f32_16x16x4_f32(
            /*neg_a=*/false, a, /*neg_b=*/false, b,
            /*c_mod=*/(short)0, acc[t], /*reuse_a=*/false, /*reuse_b=*/false);
      }
    }
  }
  // scatter D(m,n) with fp32 global atomics (segment_sum)
  const int mbase = (lane < 16) ? 0 : 8;
  const bool colok = (COUT >= NTILES * 16) ? true : (m < COUT);  // only COUT=8 masks
  if (p0 + 16 <= P) {
    // full tile: no row guards; 8 clause-able broadcast loads off one base
    const int* __restrict__ prow = pout + p0 + mbase;
    int orow[8];
#pragma unroll
    for (int r = 0; r < 8; ++r) orow[r] = prow[r];
    if (colok) {
#pragma unroll
      for (int r = 0; r < 8; ++r) {
        float* base = out + (size_t)orow[r] * COUT + m;
#pragma unroll
        for (int t = 0; t < NTILES; ++t)
          atomicAdd(base + t * 16, acc[t][r]);
      }
    }
  } else {
#pragma unroll
    for (int r = 0; r < 8; ++r) {
      const int row = mbase + r;
      if (p0 + row < P && colok) {
        float* base = out + (size_t)pout[p0 + row] * COUT + m;
#pragma unroll
        for (int t = 0; t < NTILES; ++t)
          atomicAdd(base + t * 16, acc[t][r]);
      }
    }
  }
}
#else
// Scalar fallback reading the packed-weight layout (histogram wmma==0 flags it)
template <int CIN, int COUT>
__global__ void k_sconv(const float* __restrict__ x, const v2f* __restrict__ pW,
                        const int* __restrict__ pin, const int* __restrict__ pout,
                        const int* __restrict__ kidx, float* __restrict__ out, int P) {
  constexpr int KSTEPS = CIN / 4;
  constexpr int NTILES = (COUT + 15) / 16;
  int p = blockIdx.x * blockDim.x + threadIdx.x;
  if (p >= P) return;
  const float* xr = x + (size_t)pin[p] * CIN;
  const v2f* pWk = pW + (size_t)kidx[p] * (KSTEPS * NTILES * 32);
  const int o = pout[p];
  for (int d = 0; d < COUT; ++d) {
    float s = 0.f;
    for (int c = 0; c < CIN; ++c) {
      int kk = c >> 2, rlo = c & 3;
      int laneoff = ((rlo >= 2) ? 16 : 0) + (d & 15);
      v2f e = pWk[(kk * NTILES + (d >> 4)) * 32 + laneoff];
      s += xr[c] * ((rlo & 1) ? e.y : e.x);
    }
    atomicAdd(&out[(size_t)o * COUT + d], s);
  }
}
#endif

static inline int sconv_grid(int P) {
#if HAVE_WMMA_F32X4
  int b = ((P + 15) / 16 + 7) / 8;  // 16 pairs/wave, 8 waves/block (256 thr)
#else
  int b = (P + 255) / 256;
#endif
  return b > 0 ? b : 1;
}

extern "C" void kernel_launch(void* const* d_in, const int* in_sizes, int n_in,
                              void* d_out, int out_size, void* d_ws, size_t ws_size,
                              hipStream_t stream) {
  (void)n_in; (void)out_size; (void)ws_size;
  const float* feats = (const float*)d_in[0];
  const int*   p2v   = (const int*)d_in[1];
  const int *in0 = (const int*)d_in[2], *out0 = (const int*)d_in[3], *k0 = (const int*)d_in[4];
  const int *ind = (const int*)d_in[5], *outd = (const int*)d_in[6], *kd = (const int*)d_in[7];
  const int *in1 = (const int*)d_in[8], *out1 = (const int*)d_in[9], *k1 = (const int*)d_in[10];
  const float *W0   = (const float*)d_in[11], *Wb1a = (const float*)d_in[12];
  const float *Wb1b = (const float*)d_in[13], *Wd   = (const float*)d_in[14];
  const float *Wb2a = (const float*)d_in[15], *Wb2b = (const float*)d_in[16];
  const float *Wu   = (const float*)d_in[17], *Wc   = (const float*)d_in[18];
  const float *Wfin = (const float*)d_in[19];
  const int M0 = in_sizes[0];   // feats is [M0,1]
  const int N  = in_sizes[1];   // points
  const int P0 = in_sizes[2];   // L0 map pairs
  const int Pd = in_sizes[5];   // down map pairs
  const int P1 = in_sizes[8];   // L1 map pairs
  // m1 (d_in[20]) never needed on host: scatter indices are already < m1,
  // and we zero/relu M1-sized buffers over their M0-sized upper bound.

  // ---- workspace arena (floats) ----
  float* ws   = (float*)d_ws;
  float* bufA = ws;                   // M0*8  : x   -> x0
  float* bufB = ws + (size_t)M0 * 8;  // M0*8  : y8  -> z
  float* bufC = ws + (size_t)M0 * 16; // M0*8  : block1 tmp
  float* bufD = ws + (size_t)M0 * 24; // M0*32 : x1
  float* bufE = ws + (size_t)M0 * 56; // M0*32 : y32 -> xu
  float* bufF = ws + (size_t)M0 * 88; // M0*32 : block2 tmp
  float* bufG = ws + (size_t)M0 * 120;// M0*40 : concat
  // packed weights after feature buffers (v2f = 2 floats per fragment elem)
  size_t woff = (size_t)M0 * 160;
  const int F_B1 = 81 * 2 * 1 * 32;   // 8->8   : KSTEPS=2, NTILES=1
  const int F_D  = 16 * 2 * 2 * 32;   // 8->32  : KSTEPS=2, NTILES=2
  const int F_B2 = 81 * 8 * 2 * 32;   // 32->32 : KSTEPS=8, NTILES=2
  const int F_U  = 16 * 8 * 2 * 32;   // 32->32 (up)
  const int F_C  = 81 * 10 * 1 * 32;  // 40->8  : KSTEPS=10, NTILES=1
  v2f* pWb1a = (v2f*)(ws + woff);               woff += (size_t)F_B1 * 2;
  v2f* pWb1b = (v2f*)(ws + woff);               woff += (size_t)F_B1 * 2;
  v2f* pWd   = (v2f*)(ws + woff);               woff += (size_t)F_D * 2;
  v2f* pWb2a = (v2f*)(ws + woff);               woff += (size_t)F_B2 * 2;
  v2f* pWb2b = (v2f*)(ws + woff);               woff += (size_t)F_B2 * 2;
  v2f* pWu   = (v2f*)(ws + woff);               woff += (size_t)F_U * 2;
  v2f* pWc   = (v2f*)(ws + woff);               woff += (size_t)F_C * 2;

  const int TB = 256;
  auto g1 = [](int n) { int b = (n + 255) / 256; return b > 0 ? b : 1; };
#define F4(ptr) ((float4*)(ptr))

  // ---- repack weights into WMMA B-fragment order (cheap, every call) ----
  k_packW<8, 8><<<g1(F_B1), TB, 0, stream>>>(Wb1a, pWb1a, F_B1);
  k_packW<8, 8><<<g1(F_B1), TB, 0, stream>>>(Wb1b, pWb1b, F_B1);
  k_packW<8, 32><<<g1(F_D), TB, 0, stream>>>(Wd, pWd, F_D);
  k_packW<32, 32><<<g1(F_B2), TB, 0, stream>>>(Wb2a, pWb2a, F_B2);
  k_packW<32, 32><<<g1(F_B2), TB, 0, stream>>>(Wb2b, pWb2b, F_B2);
  k_packW<32, 32><<<g1(F_U), TB, 0, stream>>>(Wu, pWu, F_U);
  k_packW<40, 8><<<g1(F_C), TB, 0, stream>>>(Wc, pWc, F_C);

  // ---- stem: 1 -> 8, relu ----
  k_zero4<<<g1(M0 * 2), TB, 0, stream>>>(F4(bufA), M0 * 2);
  k_stem<<<g1(P0), TB, 0, stream>>>(feats, W0, in0, out0, k0, bufA, P0);
  k_relu4<<<g1(M0 * 2), TB, 0, stream>>>(F4(bufA), M0 * 2);
  // ---- BasicBlock @ L0: x0 = relu(x + conv(relu(conv(x)))) ----
  k_zero4<<<g1(M0 * 2), TB, 0, stream>>>(F4(bufB), M0 * 2);
  k_sconv<8, 8><<<sconv_grid(P0), TB, 0, stream>>>(bufA, pWb1a, in0, out0, k0, bufB, P0);
  k_relu4<<<g1(M0 * 2), TB, 0, stream>>>(F4(bufB), M0 * 2);
  k_zero4<<<g1(M0 * 2), TB, 0, stream>>>(F4(bufC), M0 * 2);
  k_sconv<8, 8><<<sconv_grid(P0), TB, 0, stream>>>(bufB, pWb1b, in0, out0, k0, bufC, P0);
  k_add_relu4<<<g1(M0 * 2), TB, 0, stream>>>(F4(bufA), F4(bufA), F4(bufC), M0 * 2);  // x0
  // ---- down: 8 -> 32 (k2 s2), relu ----
  k_zero4<<<g1(M0 * 8), TB, 0, stream>>>(F4(bufD), M0 * 8);
  k_sconv<8, 32><<<sconv_grid(Pd), TB, 0, stream>>>(bufA, pWd, ind, outd, kd, bufD, Pd);
  k_relu4<<<g1(M0 * 8), TB, 0, stream>>>(F4(bufD), M0 * 8);
  // ---- BasicBlock @ L1 ----
  k_zero4<<<g1(M0 * 8), TB, 0, stream>>>(F4(bufE), M0 * 8);
  k_sconv<32, 32><<<sconv_grid(P1), TB, 0, stream>>>(bufD, pWb2a, in1, out1, k1, bufE, P1);
  k_relu4<<<g1(M0 * 8), TB, 0, stream>>>(F4(bufE), M0 * 8);
  k_zero4<<<g1(M0 * 8), TB, 0, stream>>>(F4(bufF), M0 * 8);
  k_sconv<32, 32><<<sconv_grid(P1), TB, 0, stream>>>(bufE, pWb2b, in1, out1, k1, bufF, P1);
  k_add_relu4<<<g1(M0 * 8), TB, 0, stream>>>(F4(bufD), F4(bufD), F4(bufF), M0 * 8);  // x1
  // ---- transposed conv up: swap in/out of downsample map ----
  k_zero4<<<g1(M0 * 8), TB, 0, stream>>>(F4(bufE), M0 * 8);
  k_sconv<32, 32><<<sconv_grid(Pd), TB, 0, stream>>>(bufD, pWu, outd, ind, kd, bufE, Pd);
  k_relu4<<<g1(M0 * 8), TB, 0, stream>>>(F4(bufE), M0 * 8);  // xu
  // ---- concat -> 40ch classifier conv -> relu ----
  k_concat4<<<g1(M0 * 10), TB, 0, stream>>>(F4(bufG), F4(bufE), F4(bufA), M0 * 10);
  k_zero4<<<g1(M0 * 2), TB, 0, stream>>>(F4(bufB), M0 * 2);
  k_sconv<40, 8><<<sconv_grid(P0), TB, 0, stream>>>(bufG, pWc, in0, out0, k0, bufB, P0);
  k_relu4<<<g1(M0 * 2), TB, 0, stream>>>(F4(bufB), M0 * 2);
  // ---- per-point logits ----
  k_final<<<g1(N), TB, 0, stream>>>(bufB, Wfin, p2v, (float*)d_out, N);
#undef F4
}